// ChebBen2_71159018160654
// MI455X (gfx1250) — compile-verified
//
#include <hip/hip_runtime.h>
#include <hip/hip_bf16.h>

// ChebConv x2 for MI455X (gfx1250, wave32).
//  - SpMM (dominant cost): 16 threads/edge, float4 gathers, f32 global atomics.
//    All operands fit in the 192MB L2, so atomics resolve at L2 speed.
//  - Dense x@W GEMMs: V_WMMA_F32_16X16X4_F32 (full fp32, matches reference math).
//  - Tx2 = 2*prop(Tx1) - Tx0 fused into the GEMM A-fragment load (no extra pass).

#define FDIM 64

typedef __attribute__((ext_vector_type(2))) float v2f;
typedef __attribute__((ext_vector_type(8))) float v8f;

// ---------------- degree / norm ----------------

__global__ void deg_kernel(const int* __restrict__ row, float* __restrict__ deg, int E) {
    int t = blockIdx.x * blockDim.x + threadIdx.x;
    if (t < E) atomicAdd(&deg[row[t]], 1.0f);
}

__global__ void dis_kernel(float* __restrict__ deg, int N) {
    int i = blockIdx.x * blockDim.x + threadIdx.x;
    if (i < N) {
        float d = deg[i];
        deg[i] = (d > 0.0f) ? rsqrtf(fmaxf(d, 1.0f)) : 0.0f;
    }
}

__global__ void norm_kernel(const int* __restrict__ row, const int* __restrict__ col,
                            const float* __restrict__ dis, float* __restrict__ nrm, int E) {
    int e = blockIdx.x * blockDim.x + threadIdx.x;
    if (e < E) nrm[e] = -dis[row[e]] * dis[col[e]];
}

// ---------------- SpMM: out[row] += norm * h[col]  (out pre-zeroed) ----------------
// 16 threads per edge, 4 floats each: coalesced 64-float gather per edge.

__global__ __launch_bounds__(256) void prop_kernel(const float* __restrict__ h,
                                                   const int* __restrict__ row,
                                                   const int* __restrict__ col,
                                                   const float* __restrict__ nrm,
                                                   float* __restrict__ out, int E) {
    int t = blockIdx.x * blockDim.x + threadIdx.x;
    if (t >= E * 16) return;
    int e = t >> 4;
    int q = t & 15;
    float s = nrm[e];
    int r = row[e];
    int c = col[e];
    float4 v = *(const float4*)(h + (size_t)c * FDIM + q * 4);
    float* o = out + (size_t)r * FDIM + q * 4;
    atomicAdd(o + 0, s * v.x);
    atomicAdd(o + 1, s * v.y);
    atomicAdd(o + 2, s * v.z);
    atomicAdd(o + 3, s * v.w);
}

// ---------------- fused Chebyshev GEMM ----------------
// out = X0@W[0] + X1@W[1] + (2*P2 - X0)@W[2] + bias, optional ReLU.
// One wave per 16-row tile; NT column tiles of 16. fp32 WMMA 16x16x4.
//
// Fragment layouts (ISA 7.12.2, wave32):
//   A 16x4 f32 : lane l (l<16): {A[l][kb], A[l][kb+1]}, l>=16: K base +2.
//   B 4x16 f32 : lane l (l<16): {B[kb][l], B[kb+1][l]}, l>=16: K base +2, N=l-16.
//   C/D 16x16  : VGPR r -> row (r + 8*(l>=16)), col = l&15.

template <int NOUT, bool RELU>
__global__ __launch_bounds__(256) void cheb_gemm_kernel(const float* __restrict__ X0,
                                                        const float* __restrict__ X1,
                                                        const float* __restrict__ P2,
                                                        const float* __restrict__ W,   // [3][64][NOUT]
                                                        const float* __restrict__ bias,
                                                        float* __restrict__ out, int Nrows) {
    constexpr int NT = (NOUT + 15) / 16;
    int wave = threadIdx.x >> 5;
    int lane = threadIdx.x & 31;
    int tile = blockIdx.x * (blockDim.x >> 5) + wave;
    int m0 = tile * 16;
    if (m0 >= Nrows) return;                 // wave-uniform: EXEC stays all-1s for WMMA

    int half = lane >> 4;                    // which K-pair of the x4 step
    int lm = lane & 15;
    int M = m0 + lm;
    int Ml = (M < Nrows) ? M : (Nrows - 1);  // clamped load row (stores are guarded)

    v8f acc[NT];
#pragma unroll
    for (int i = 0; i < NT; ++i) acc[i] = (v8f)0.0f;

#pragma unroll
    for (int k = 0; k < 3; ++k) {
        const float* A = (k == 0) ? X0 : (k == 1) ? X1 : P2;
#pragma unroll
        for (int kg = 0; kg < FDIM / 4; ++kg) {
            int kb = kg * 4 + half * 2;
            float a0 = A[(size_t)Ml * FDIM + kb];
            float a1 = A[(size_t)Ml * FDIM + kb + 1];
            if (k == 2) {  // Tx2 = 2*P2 - Tx0, fused
                a0 = 2.0f * a0 - X0[(size_t)Ml * FDIM + kb];
                a1 = 2.0f * a1 - X0[(size_t)Ml * FDIM + kb + 1];
            }
            v2f av;
            av.x = a0;
            av.y = a1;
#pragma unroll
            for (int nt = 0; nt < NT; ++nt) {
                int n = nt * 16 + lm;
                int nc = (n < NOUT) ? n : 0;  // clamp address, zero value below (no branch)
                float b0 = W[k * FDIM * NOUT + kb * NOUT + nc];
                float b1 = W[k * FDIM * NOUT + (kb + 1) * NOUT + nc];
                if (n >= NOUT) { b0 = 0.0f; b1 = 0.0f; }
                v2f bv;
                bv.x = b0;
                bv.y = b1;
                acc[nt] = __builtin_amdgcn_wmma_f32_16x16x4_f32(
                    false, av, false, bv, (short)0, acc[nt], false, false);
            }
        }
    }

#pragma unroll
    for (int nt = 0; nt < NT; ++nt) {
        int n = nt * 16 + lm;
        if (n < NOUT) {
            float bb = bias[n];
#pragma unroll
            for (int r = 0; r < 8; ++r) {
                int rr = m0 + r + half * 8;
                if (rr < Nrows) {
                    float v = acc[nt][r] + bb;
                    if (RELU) v = fmaxf(v, 0.0f);
                    out[(size_t)rr * NOUT + n] = v;
                }
            }
        }
    }
}

// ---------------- launcher ----------------

extern "C" void kernel_launch(void* const* d_in, const int* in_sizes, int n_in,
                              void* d_out, int out_size, void* d_ws, size_t ws_size,
                              hipStream_t stream) {
    const float* x  = (const float*)d_in[0];
    const int*   ei = (const int*)d_in[1];   // edge_index [2][E]
    const float* W1 = (const float*)d_in[2];
    const float* b1 = (const float*)d_in[3];
    const float* W2 = (const float*)d_in[4];
    const float* b2 = (const float*)d_in[5];
    float* out = (float*)d_out;

    const int N = in_sizes[0] / FDIM;   // 100000
    const int E = in_sizes[1] / 2;      // 1600000
    const int* row = ei;
    const int* col = ei + E;

    auto align256 = [](size_t v) { return (v + 255) & ~(size_t)255; };
    char* w = (char*)d_ws;
    float* dis  = (float*)w; w += align256((size_t)N * 4);
    float* nrm  = (float*)w; w += align256((size_t)E * 4);
    float* bufA = (float*)w; w += align256((size_t)N * FDIM * 4);
    float* bufB = (float*)w; w += align256((size_t)N * FDIM * 4);
    float* hbuf = (float*)w; w += align256((size_t)N * FDIM * 4);
    (void)ws_size;

    const int B = 256;
    const size_t featBytes = (size_t)N * FDIM * 4;
    const int edgeBlocks = (E + B - 1) / B;
    const int propBlocks = (int)(((size_t)E * 16 + B - 1) / B);
    const int tiles = (N + 15) / 16;
    const int gemmBlocks = (tiles + 7) / 8;   // 8 waves (16-row tiles) per 256-thread block

    // normalization
    hipMemsetAsync(dis, 0, (size_t)N * 4, stream);
    deg_kernel<<<edgeBlocks, B, 0, stream>>>(row, dis, E);
    dis_kernel<<<(N + B - 1) / B, B, 0, stream>>>(dis, N);
    norm_kernel<<<edgeBlocks, B, 0, stream>>>(row, col, dis, nrm, E);

    // layer 1: P1 = prop(x), P2 = prop(P1), h = relu(cheb(x, P1, P2; W1, b1))
    hipMemsetAsync(bufA, 0, featBytes, stream);
    prop_kernel<<<propBlocks, B, 0, stream>>>(x, row, col, nrm, bufA, E);
    hipMemsetAsync(bufB, 0, featBytes, stream);
    prop_kernel<<<propBlocks, B, 0, stream>>>(bufA, row, col, nrm, bufB, E);
    cheb_gemm_kernel<64, true><<<gemmBlocks, B, 0, stream>>>(x, bufA, bufB, W1, b1, hbuf, N);

    // layer 2: reuse scratch; out = cheb(h, prop(h), prop(prop(h)); W2, b2)
    hipMemsetAsync(bufA, 0, featBytes, stream);
    prop_kernel<<<propBlocks, B, 0, stream>>>(hbuf, row, col, nrm, bufA, E);
    hipMemsetAsync(bufB, 0, featBytes, stream);
    prop_kernel<<<propBlocks, B, 0, stream>>>(bufA, row, col, nrm, bufB, E);
    cheb_gemm_kernel<40, false><<<gemmBlocks, B, 0, stream>>>(hbuf, bufA, bufB, W2, b2, out, N);
}